// GritLMPooler_15882789060666
// MI455X (gfx1250) — compile-verified
//
#include <hip/hip_runtime.h>
#include <hip/hip_bf16.h>
#include <math.h>

// ---------------------------------------------------------------------------
// GritLM pooler for MI455X (gfx1250, wave32).
//
// hidden_states: [T, D] fp32 packed ragged batch (T = 65536, D = 4096 -> 1 GB)
// prompt_lens  : [B] int32 (sums to T)
// instruction_lens : [B] int32
// out          : [B, D] fp32, L2-normalized masked mean per sequence.
//
// Roofline: ~1 GB read @ 23.3 TB/s => ~46 us floor; compute negligible.
// All-fp32 path using V_WMMA_F32_16X16X4_F32 to fuse the 4-token reduction:
//   A (16x4) = all-ones (identical rows), B (4x16) = 4 tokens x 16 columns,
//   D = A*B + C accumulates per-column sums in f32 (exact).
// The instruction-prefix mask is a *contiguous* suffix predicate per sequence,
// so instead of multiplying by 0/1 we simply start the token loop at
// kStart = max(chunkStart, segStart + ilen): no mask VALU in the hot loop and
// the masked-out rows are never even loaded (~3% HBM traffic saved).
// Streaming reads use non-temporal hints (1 GB >> 192 MB L2, zero reuse).
// ---------------------------------------------------------------------------

typedef __attribute__((ext_vector_type(2))) float v2f;
typedef __attribute__((ext_vector_type(8))) float v8f;

#define D_HIDDEN   4096
#define ROW_CHUNKS 4      // row-chunk splits per sequence (partial sums)
#define WAVES_PER_BLOCK 8
#define COLS_PER_WAVE   16
#define COLS_PER_BLOCK  (WAVES_PER_BLOCK * COLS_PER_WAVE)   // 128
#define EPS_NORM   1e-12f

// ---------------------------------------------------------------------------
// Kernel 1: partial column sums of kept tokens via WMMA f32 16x16x4.
// grid = (D/128, ROW_CHUNKS, B), block = 256 (8 wave32).
// partial[b][r][c] = sum over kept tokens in row-chunk r of seq b of hs[t][c]
// ---------------------------------------------------------------------------
__global__ __launch_bounds__(256)
void gritlm_partial_sums_wmma(const float* __restrict__ hs,
                              const int*   __restrict__ plens,
                              const int*   __restrict__ ilens,
                              float*       __restrict__ partial,
                              int B, long long T) {
    const int tid   = threadIdx.x;
    const int wave  = tid >> 5;          // 0..7
    const int lane  = tid & 31;          // wave32
    const int half  = lane >> 4;         // 0: lanes 0-15, 1: lanes 16-31
    const int lan16 = lane & 15;

    const int colBlock = blockIdx.x;
    const int r        = blockIdx.y;
    const int b        = blockIdx.z;

    // Ragged segment offsets (B tiny; uniform scalar loop -> SMEM loads).
    long long segStart = 0;
    for (int i = 0; i < b; ++i) segStart += (long long)plens[i];
    const int  plen = plens[b];
    const int  ilen = ilens[b];
    const long long segEnd = segStart + plen;

    const int chunkLen = (plen + ROW_CHUNKS - 1) / ROW_CHUNKS;
    long long chunkStart = segStart + (long long)r * chunkLen;
    long long chunkEnd   = chunkStart + chunkLen;
    if (chunkEnd > segEnd) chunkEnd = segEnd;

    // First kept token overall; loop region is [kStart, chunkEnd).
    long long kStart = segStart + (long long)(ilen < plen ? ilen : plen);
    if (kStart < chunkStart) kStart = chunkStart;

    const int colBase = colBlock * COLS_PER_BLOCK + wave * COLS_PER_WAVE;
    const int cb      = colBase + lan16;       // column this lane loads

    // B-operand row offsets for this lane (K layout: VGPR0 -> K=0 / K=2,
    // VGPR1 -> K=1 / K=3 across the two lane halves).
    const int rowOff0 = 2 * half;       // token t0 (half 0) / t2 (half 1)
    const int rowOff1 = 1 + 2 * half;   // token t1 (half 0) / t3 (half 1)

    v8f acc = {};                       // C/D accumulator (all rows identical)
    v2f Aones; Aones.x = 1.0f; Aones.y = 1.0f;

    const long long nTok  = chunkEnd > kStart ? (chunkEnd - kStart) : 0;
    const long long nFull = nTok & ~3LL;       // full groups of 4 tokens

    const float* p0 = hs + (kStart + rowOff0) * (long long)D_HIDDEN + cb;
    const float* p1 = hs + (kStart + rowOff1) * (long long)D_HIDDEN + cb;
    const long long step = 4LL * D_HIDDEN;

    // Hot loop: 2 coalesced non-temporal b32 loads + 1 v_wmma_f32_16x16x4_f32
    // per wave per 4 tokens x 16 cols (256 B). Wave-uniform trip count ->
    // EXEC stays all-1s (WMMA requirement). No mask math in the loop.
    #pragma unroll 8
    for (long long i = 0; i < nFull; i += 4) {
        v2f Bm;
        Bm.x = __builtin_nontemporal_load(p0);
        Bm.y = __builtin_nontemporal_load(p1);
        acc = __builtin_amdgcn_wmma_f32_16x16x4_f32(
            false, Aones, false, Bm, (short)0, acc, false, false);
        p0 += step; p1 += step;
    }

    // Ragged tail (1-3 tokens), uniform branch; range mask in A, clamped rows.
    if (nFull < nTok) {
        const long long t0   = kStart + nFull;
        const long long row0 = t0 + rowOff0;
        const long long row1 = t0 + rowOff1;
        v2f A;
        A.x = (row0 < chunkEnd) ? 1.0f : 0.0f;
        A.y = (row1 < chunkEnd) ? 1.0f : 0.0f;
        const long long r0c = row0 < T - 1 ? row0 : T - 1;
        const long long r1c = row1 < T - 1 ? row1 : T - 1;
        v2f Bm;
        Bm.x = __builtin_nontemporal_load(hs + r0c * (long long)D_HIDDEN + cb);
        Bm.y = __builtin_nontemporal_load(hs + r1c * (long long)D_HIDDEN + cb);
        acc = __builtin_amdgcn_wmma_f32_16x16x4_f32(
            false, A, false, Bm, (short)0, acc, false, false);
    }

    // D row 0 (acc[0], lanes 0-15) holds the per-column sums.
    if (lane < 16) {
        partial[((long long)b * ROW_CHUNKS + r) * D_HIDDEN + colBase + lane] = acc[0];
    }
}

// ---------------------------------------------------------------------------
// Kernel 2: combine partials -> mean -> L2 normalize. One block per sequence.
// Fixed reduction order => deterministic.
// ---------------------------------------------------------------------------
__global__ __launch_bounds__(256)
void gritlm_finalize(const float* __restrict__ partial,
                     const int*   __restrict__ plens,
                     const int*   __restrict__ ilens,
                     float*       __restrict__ out,
                     int B) {
    const int b   = blockIdx.x;
    const int tid = threadIdx.x;
    const float count = (float)(plens[b] - ilens[b]);
    const float invCount = 1.0f / count;

    const int nIter = D_HIDDEN / 256;   // 16 columns per thread
    float mean[nIter];
    float sumsq = 0.0f;

    #pragma unroll
    for (int i = 0; i < nIter; ++i) {
        const int c = tid + i * 256;
        float s = 0.0f;
        #pragma unroll
        for (int r = 0; r < ROW_CHUNKS; ++r)
            s += partial[((long long)b * ROW_CHUNKS + r) * D_HIDDEN + c];
        float m = s * invCount;
        mean[i] = m;
        sumsq += m * m;
    }

    __shared__ float red[256];
    red[tid] = sumsq;
    __syncthreads();
    #pragma unroll
    for (int s = 128; s > 0; s >>= 1) {
        if (tid < s) red[tid] += red[tid + s];
        __syncthreads();
    }
    __shared__ float invNorm;
    if (tid == 0) {
        float n = sqrtf(red[0]);
        invNorm = 1.0f / fmaxf(n, EPS_NORM);
    }
    __syncthreads();
    const float scale = invNorm;

    #pragma unroll
    for (int i = 0; i < nIter; ++i) {
        const int c = tid + i * 256;
        out[(long long)b * D_HIDDEN + c] = mean[i] * scale;
    }
}

// ---------------------------------------------------------------------------
extern "C" void kernel_launch(void* const* d_in, const int* in_sizes, int n_in,
                              void* d_out, int out_size, void* d_ws, size_t ws_size,
                              hipStream_t stream) {
    const float* hs    = (const float*)d_in[0];
    const int*   plens = (const int*)d_in[1];
    const int*   ilens = (const int*)d_in[2];
    float*       out   = (float*)d_out;

    const int B = in_sizes[1];
    const long long T = (long long)in_sizes[0] / D_HIDDEN;

    float* partial = (float*)d_ws;   // [B][ROW_CHUNKS][D] fp32 (2 MB for B=32)

    dim3 grid1(D_HIDDEN / COLS_PER_BLOCK, ROW_CHUNKS, B);
    gritlm_partial_sums_wmma<<<grid1, 256, 0, stream>>>(hs, plens, ilens,
                                                        partial, B, T);

    gritlm_finalize<<<B, 256, 0, stream>>>(partial, plens, ilens, out, B);
}